// IWeightedSymmetricTensorProduct_46497315947113
// MI455X (gfx1250) — compile-verified
//
#include <hip/hip_runtime.h>
#include <hip/hip_bf16.h>

typedef __attribute__((ext_vector_type(2))) float v2f;
typedef __attribute__((ext_vector_type(8))) float v8f;

#define UDIM 128
#define NIN 4
#define NX1 4
#define NOUT 4
#define KPAD 36          // 34 monomials padded to 9 chunks of K=4
#define ROWS_PER_BLOCK 16

// monomial id layout: [0..3] deg1, [4..13] deg2 (i<=j), [14..33] deg3 (i<=j<=k)
__host__ __device__ constexpr int pair_id(int i, int j) {
  return 4 + i * 4 - (i * (i - 1)) / 2 + (j - i);
}
__host__ __device__ constexpr int trip_base(int i) {
  return (i == 0) ? 0 : (i == 1) ? 10 : (i == 2) ? 16 : 19;
}
__host__ __device__ constexpr int trip_id(int i, int j, int k) {
  int s = 0;
  for (int b = i; b < j; ++b) s += (4 - b);
  return 14 + trip_base(i) + s + (k - j);
}

// ---- preprocessing: fold 104 runtime paths into dense W[16][KPAD] ----
__global__ void build_w_kernel(const int* __restrict__ in1, const int* __restrict__ xi1,
                               const int* __restrict__ o1, const float* __restrict__ c1, int P1,
                               const int* __restrict__ in2, const int* __restrict__ xi2,
                               const int* __restrict__ o2, const float* __restrict__ c2, int P2,
                               const int* __restrict__ in3, const int* __restrict__ xi3,
                               const int* __restrict__ o3, const float* __restrict__ c3, int P3,
                               float* __restrict__ W) {
  if (threadIdx.x != 0 || blockIdx.x != 0) return;
  for (int t = 0; t < 16 * KPAD; ++t) W[t] = 0.f;
  for (int p = 0; p < P1; ++p) {
    int mono = xi1[p];  // degree-1 monomial id == x1 index
    W[(o1[p] * NIN + in1[p]) * KPAD + mono] += c1[p];
  }
  for (int p = 0; p < P2; ++p) {
    int j0 = xi2[2 * p], j1 = xi2[2 * p + 1];  // sorted
    W[(o2[p] * NIN + in2[p]) * KPAD + pair_id(j0, j1)] += c2[p];
  }
  for (int p = 0; p < P3; ++p) {
    int j0 = xi3[3 * p], j1 = xi3[3 * p + 1], j2 = xi3[3 * p + 2];  // sorted
    W[(o3[p] * NIN + in3[p]) * KPAD + trip_id(j0, j1, j2)] += c3[p];
  }
}

// ---- main kernel: 16 points per wave, P = W*m via 9 chained WMMA f32 16x16x4 ----
__global__ void __launch_bounds__(256)
stp_kernel(const float* __restrict__ x0, const int* __restrict__ i0,
           const float* __restrict__ x1, const float* __restrict__ W,
           float* __restrict__ out, int B) {
  const int tid  = threadIdx.x;
  const int wave = tid >> 5;
  const int l15  = tid & 15;          // point index within wave (N), also matrix row for A
  const bool hi  = (tid & 16) != 0;   // lanes 16..31 carry K+2/K+3 halves and rows 8..15
  const int u    = wave * 16 + l15;   // 8 waves * 16 = 128 channels per block

  // Wave-resident A fragments: A[M=l15][K] weight chunks, 2 VGPRs per chunk.
  v2f Aw[9];
#pragma unroll
  for (int kk = 0; kk < 9; ++kk) {
    const int kb = kk * 4 + (hi ? 2 : 0);
    Aw[kk] = *reinterpret_cast<const v2f*>(W + l15 * KPAD + kb);
  }

  const int b0 = blockIdx.x * ROWS_PER_BLOCK;
#pragma unroll 1
  for (int r = 0; r < ROWS_PER_BLOCK; ++r) {
    const int b = b0 + r;
    if (b >= B) break;

    const float* xr = x1 + (size_t)b * (NX1 * UDIM);
    const float xv0 = xr[0 * UDIM + u];
    const float xv1 = xr[1 * UDIM + u];
    const float xv2 = xr[2 * UDIM + u];
    const float xv3 = xr[3 * UDIM + u];

    const int row = i0[b];
    const float* x0r = x0 + (size_t)row * (NIN * UDIM);
    const float a0 = x0r[0 * UDIM + u];
    const float a1 = x0r[1 * UDIM + u];
    const float a2 = x0r[2 * UDIM + u];
    const float a3 = x0r[3 * UDIM + u];

    // 34 monomials, straight-line scalar code (literal names only => stays in VGPRs)
    const float m0 = xv0, m1 = xv1, m2 = xv2, m3 = xv3;
    const float m4 = xv0 * xv0, m5 = xv0 * xv1, m6 = xv0 * xv2, m7 = xv0 * xv3;
    const float m8 = xv1 * xv1, m9 = xv1 * xv2, m10 = xv1 * xv3;
    const float m11 = xv2 * xv2, m12 = xv2 * xv3, m13 = xv3 * xv3;
    const float m14 = m4 * xv0, m15 = m4 * xv1, m16 = m4 * xv2, m17 = m4 * xv3;
    const float m18 = m5 * xv1, m19 = m5 * xv2, m20 = m5 * xv3;
    const float m21 = m6 * xv2, m22 = m6 * xv3;
    const float m23 = m7 * xv3;
    const float m24 = m8 * xv1, m25 = m8 * xv2, m26 = m8 * xv3;
    const float m27 = m9 * xv2, m28 = m9 * xv3;
    const float m29 = m10 * xv3;
    const float m30 = m11 * xv2, m31 = m11 * xv3;
    const float m32 = m12 * xv3;
    const float m33 = m13 * xv3;
    const float m34 = 0.f, m35 = 0.f;

    // P(16 rows x 16 pts) = sum_k A(16x4) * B(4x16), chained accumulate in C
    v8f acc = {0.f, 0.f, 0.f, 0.f, 0.f, 0.f, 0.f, 0.f};
    v2f bm;
#define STP_WMMA(KK, E0, E1, E2, E3)                                            \
    bm.x = hi ? (E2) : (E0);                                                    \
    bm.y = hi ? (E3) : (E1);                                                    \
    acc = __builtin_amdgcn_wmma_f32_16x16x4_f32(false, Aw[KK], false, bm,       \
                                                (short)0, acc, false, false);
    STP_WMMA(0, m0,  m1,  m2,  m3)
    STP_WMMA(1, m4,  m5,  m6,  m7)
    STP_WMMA(2, m8,  m9,  m10, m11)
    STP_WMMA(3, m12, m13, m14, m15)
    STP_WMMA(4, m16, m17, m18, m19)
    STP_WMMA(5, m20, m21, m22, m23)
    STP_WMMA(6, m24, m25, m26, m27)
    STP_WMMA(7, m28, m29, m30, m31)
    STP_WMMA(8, m32, m33, m34, m35)
#undef STP_WMMA

    // lane holds rows M = 8*hi + [0..7] of point l15: combine with x0 gather
    const float o0 = a0 * acc[0] + a1 * acc[1] + a2 * acc[2] + a3 * acc[3];
    const float o1 = a0 * acc[4] + a1 * acc[5] + a2 * acc[6] + a3 * acc[7];

    float* orow = out + (size_t)b * (NOUT * UDIM);
    const int nb = hi ? 2 : 0;
    orow[(nb + 0) * UDIM + u] = o0;
    orow[(nb + 1) * UDIM + u] = o1;
  }
}

extern "C" void kernel_launch(void* const* d_in, const int* in_sizes, int n_in,
                              void* d_out, int out_size, void* d_ws, size_t ws_size,
                              hipStream_t stream) {
  const float* x0 = (const float*)d_in[0];
  const int*   i0 = (const int*)d_in[1];
  const float* x1 = (const float*)d_in[2];
  const int*   in1 = (const int*)d_in[3];
  const int*   xi1 = (const int*)d_in[4];
  const int*   o1  = (const int*)d_in[5];
  const float* c1  = (const float*)d_in[6];
  const int*   in2 = (const int*)d_in[7];
  const int*   xi2 = (const int*)d_in[8];
  const int*   o2  = (const int*)d_in[9];
  const float* c2  = (const float*)d_in[10];
  const int*   in3 = (const int*)d_in[11];
  const int*   xi3 = (const int*)d_in[12];
  const int*   o3  = (const int*)d_in[13];
  const float* c3  = (const float*)d_in[14];

  float* W   = (float*)d_ws;
  float* out = (float*)d_out;
  const int B  = in_sizes[1];
  const int P1 = in_sizes[3];
  const int P2 = in_sizes[7];
  const int P3 = in_sizes[11];

  build_w_kernel<<<1, 1, 0, stream>>>(in1, xi1, o1, c1, P1,
                                      in2, xi2, o2, c2, P2,
                                      in3, xi3, o3, c3, P3, W);

  const int grid = (B + ROWS_PER_BLOCK - 1) / ROWS_PER_BLOCK;
  stp_kernel<<<grid, 256, 0, stream>>>(x0, i0, x1, W, out, B);
}